// HashedEmbeddingBag_61091614818812
// MI455X (gfx1250) — compile-verified
//
#include <hip/hip_runtime.h>
#include <hip/hip_bf16.h>

// ---------------------------------------------------------------------------
// HashedEmbeddingBag forward, CDNA5 (gfx1250), wave32.
//
//   out[b, d] = sum_{e<50} weight[(idx[b,e]*P1 + d*P2) mod 2e6]
//
// Two-phase:
//   Phase 1: b[bag,e] = (idx*P1) mod W  precomputed once into d_ws,
//            laid out [entry][bag] so phase 2 reads b for a 4-bag K-quad
//            as one aligned b128 load at a wave-uniform address.
//   Phase 2: one wave per 16-bag x 16-dim tile, accumulated with
//            V_WMMA_F32_16X16X4_F32 (exact f32). Per WMMA: 1 b128 load,
//            2 random 4B gathers/lane, ~4 VALU/gather (add, sub, min,
//            cndmask). Weight table (8MB) + b table (3.2MB) are L2-resident;
//            the kernel is bound by the random L2 gather stream.
// ---------------------------------------------------------------------------

typedef float v2f __attribute__((ext_vector_type(2)));
typedef float v8f __attribute__((ext_vector_type(8)));

#define EMB_DIM   128
#define W_SIZE    2000000u
#define BATCH     16384
#define BAG_LEN   50
// P1 = 9824516537       -> P1 mod W = 516537
// P2 = 57857966300227   -> P2 mod W = 300227
#define P1_MOD    516537ull
#define P2_MOD    300227u

// Branchless h = (t < W) ? t : t - W  for t in [0, 2W):  umin(t, t - W).
__device__ __forceinline__ unsigned mod_fold(unsigned t) {
    unsigned t2 = t - W_SIZE;          // wraps huge when t < W
    return (t < t2) ? t : t2;          // -> v_min_u32
}

// ---------------------------------------------------------------------------
// Phase 1: b_ws[e*BATCH + bag] = (idx[bag*BAG_LEN + e] * P1) mod W
// ---------------------------------------------------------------------------
__global__ __launch_bounds__(256)
void hbag_hash_pre(const int* __restrict__ indices, unsigned* __restrict__ bws) {
    const unsigned t = blockIdx.x * 256u + threadIdx.x;     // 0 .. 819199
    if (t < BATCH * BAG_LEN) {
        const unsigned bag = t / BAG_LEN;
        const unsigned e   = t - bag * BAG_LEN;
        const unsigned i   = (unsigned)indices[t];          // coalesced load
        const unsigned b   =
            (unsigned)(((unsigned long long)i * P1_MOD) % W_SIZE);
        bws[(size_t)e * BATCH + bag] = b;                   // scattered 4B store
    }
}

// ---------------------------------------------------------------------------
// Phase 2: WMMA-accumulated gather-sum using precomputed b table.
// ---------------------------------------------------------------------------
__global__ __launch_bounds__(32)
void HashedEmbeddingBag_61091614818812_kernel(const float*    __restrict__ weight,
                                              const unsigned* __restrict__ bws,
                                              float*          __restrict__ out) {
    const unsigned lane = threadIdx.x;        // 0..31 (wave32)
    const unsigned half = lane >> 4;          // 0: K=0/1 slots, 1: K=2/3 slots
    const unsigned l15  = lane & 15u;

    const unsigned tile    = blockIdx.x;      // 8192 tiles
    const unsigned dimTile = tile & 7u;
    const unsigned bagTile = tile >> 3u;

    const unsigned dim  = dimTile * 16u + l15;
    const unsigned bag0 = bagTile * 16u;

    // Per-dim hash offset: (d * P2) mod W, 32-bit safe (127*300227 < 2^32).
    const unsigned o = (dim * P2_MOD) % W_SIZE;

    v8f c = {0.f, 0.f, 0.f, 0.f, 0.f, 0.f, 0.f, 0.f};

    for (unsigned q = 0; q < 4u; ++q) {
        const unsigned bagQ = bag0 + 4u * q;       // K slot k -> bag (bagQ + k)

        // A[m, k] = (m == 4q + k).  16x4 f32 A layout: lanes0-15 hold K=0 (v0),
        // K=1 (v1); lanes16-31 hold K=2 (v0), K=3 (v1); M = lane&15.
        const unsigned k0 = 2u * half;
        v2f a;
        a.x = (l15 == 4u * q + k0)      ? 1.0f : 0.0f;
        a.y = (l15 == 4u * q + k0 + 1u) ? 1.0f : 0.0f;

        // Wave-uniform, 16B-aligned: b for the 4 bags of this K-quad, entry e.
        const uint4* __restrict__ bq = (const uint4*)(bws + bagQ);

#pragma unroll 10
        for (int e = 0; e < BAG_LEN; ++e) {
            const uint4 b4 = bq[(size_t)e * (BATCH / 4)];   // one b128 load

            // This lane's two B-matrix rows (bags bagQ+k0, bagQ+k0+1).
            const unsigned bx = half ? b4.z : b4.x;         // v_cndmask
            const unsigned by = half ? b4.w : b4.y;

            const unsigned hx = mod_fold(bx + o);           // add, sub, min
            const unsigned hy = mod_fold(by + o);

            // B[k, n] 4x16 f32: lanes0-15 hold K=0 (v0), K=1 (v1);
            // lanes16-31 hold K=2 (v0), K=3 (v1); N = lane&15.
            v2f bb;
            bb.x = weight[hx];                              // random L2 gather
            bb.y = weight[hy];

            c = __builtin_amdgcn_wmma_f32_16x16x4_f32(
                    /*neg_a=*/false, a, /*neg_b=*/false, bb,
                    /*c_mod=*/(short)0, c, /*reuse_a=*/false, /*reuse_b=*/false);
        }
    }

    // C/D layout: VGPR r, lanes0-15 -> M=r, lanes16-31 -> M=r+8; N = lane&15.
    float* op = out + (size_t)(bag0 + 8u * half) * EMB_DIM + dim;
#pragma unroll
    for (int r = 0; r < 8; ++r)
        __builtin_nontemporal_store(c[r], op + (size_t)r * EMB_DIM);
}

// ---------------------------------------------------------------------------
// Fallback (ws too small): single-pass, hash recomputed per dim-tile.
// ---------------------------------------------------------------------------
__global__ __launch_bounds__(32)
void hbag_single_pass(const float* __restrict__ weight,
                      const int*   __restrict__ indices,
                      float*       __restrict__ out) {
    const unsigned lane = threadIdx.x;
    const unsigned half = lane >> 4;
    const unsigned l15  = lane & 15u;
    const unsigned tile    = blockIdx.x;
    const unsigned dimTile = tile & 7u;
    const unsigned bagTile = tile >> 3u;
    const unsigned dim  = dimTile * 16u + l15;
    const unsigned bag0 = bagTile * 16u;
    const unsigned o = (dim * P2_MOD) % W_SIZE;

    v8f c = {0.f, 0.f, 0.f, 0.f, 0.f, 0.f, 0.f, 0.f};

    for (unsigned q = 0; q < 4u; ++q) {
        const unsigned bagQ = bag0 + 4u * q;
        const unsigned k0   = 2u * half;
        v2f a;
        a.x = (l15 == 4u * q + k0)      ? 1.0f : 0.0f;
        a.y = (l15 == 4u * q + k0 + 1u) ? 1.0f : 0.0f;
        const int* __restrict__ ip = indices + (size_t)bagQ * BAG_LEN;
#pragma unroll 10
        for (int e = 0; e < BAG_LEN; ++e) {
            const unsigned i0 = (unsigned)ip[e];
            const unsigned i1 = (unsigned)ip[1 * BAG_LEN + e];
            const unsigned i2 = (unsigned)ip[2 * BAG_LEN + e];
            const unsigned i3 = (unsigned)ip[3 * BAG_LEN + e];
            const unsigned b0 = (unsigned)(((unsigned long long)i0 * P1_MOD) % W_SIZE);
            const unsigned b1 = (unsigned)(((unsigned long long)i1 * P1_MOD) % W_SIZE);
            const unsigned b2 = (unsigned)(((unsigned long long)i2 * P1_MOD) % W_SIZE);
            const unsigned b3 = (unsigned)(((unsigned long long)i3 * P1_MOD) % W_SIZE);
            const unsigned bx = half ? b2 : b0;
            const unsigned by = half ? b3 : b1;
            const unsigned hx = mod_fold(bx + o);
            const unsigned hy = mod_fold(by + o);
            v2f bb;
            bb.x = weight[hx];
            bb.y = weight[hy];
            c = __builtin_amdgcn_wmma_f32_16x16x4_f32(
                    false, a, false, bb, (short)0, c, false, false);
        }
    }

    float* op = out + (size_t)(bag0 + 8u * half) * EMB_DIM + dim;
#pragma unroll
    for (int r = 0; r < 8; ++r)
        __builtin_nontemporal_store(c[r], op + (size_t)r * EMB_DIM);
}

extern "C" void kernel_launch(void* const* d_in, const int* in_sizes, int n_in,
                              void* d_out, int out_size, void* d_ws, size_t ws_size,
                              hipStream_t stream) {
    const float* weight  = (const float*)d_in[0];   // [2,000,000] f32
    const int*   indices = (const int*)d_in[1];     // [16384 * 50] int
    float*       out     = (float*)d_out;           // [16384 * 128] f32

    const unsigned tiles   = (BATCH / 16) * (EMB_DIM / 16);   // 8192
    const size_t   ws_need = (size_t)BATCH * BAG_LEN * sizeof(unsigned);

    if (ws_size >= ws_need) {
        unsigned* bws = (unsigned*)d_ws;
        const unsigned n1 = BATCH * BAG_LEN;                  // 819200
        hbag_hash_pre<<<(n1 + 255) / 256, 256, 0, stream>>>(indices, bws);
        HashedEmbeddingBag_61091614818812_kernel<<<tiles, 32, 0, stream>>>(
            weight, bws, out);
    } else {
        hbag_single_pass<<<tiles, 32, 0, stream>>>(weight, indices, out);
    }
}